// PerceptualEntropy_54082228191888
// MI455X (gfx1250) — compile-verified
//
#include <hip/hip_runtime.h>
#include <hip/hip_bf16.h>

#define NBANDS 24
#define FBINS 512
#define ROWS_PER_BLOCK 16
#define BLOCK_THREADS 512   // 16 wave32s; one wave per (b,t) row

__device__ __constant__ int d_widths[NBANDS] = {
    3, 3, 3, 3, 4, 4, 4, 5, 5, 6, 7, 8, 9, 10, 12, 14, 18, 22, 29, 35, 42, 58, 80, 128};

typedef __attribute__((ext_vector_type(2))) float v2f;
typedef __attribute__((ext_vector_type(8))) float v8f;

__global__ __launch_bounds__(BLOCK_THREADS)
void pe_main(const float* __restrict__ lm, const float* __restrict__ re,
             const float* __restrict__ im, const float* __restrict__ S,
             float* __restrict__ partials, int nrows)
{
    __shared__ float psumL[ROWS_PER_BLOCK][NBANDS];
    __shared__ float lsumL[ROWS_PER_BLOCK][NBANDS];
    __shared__ float sL[NBANDS][32];              // spread_function, cols zero-padded to 32
    __shared__ float spreadL[ROWS_PER_BLOCK][NBANDS];
    __shared__ float tL[ROWS_PER_BLOCK][NBANDS];
    __shared__ float ctestL[NBANDS];
    __shared__ float invkL[NBANDS];
    __shared__ float4 binTabL[FBINS];             // {tq, 6/k, wnorm/k, (float)seg}
    __shared__ unsigned char segL[FBINS];
    __shared__ float waveAcc[ROWS_PER_BLOCK];
    __shared__ __align__(16) float reL[ROWS_PER_BLOCK][FBINS];  // async-staged real row
    __shared__ __align__(16) float imL[ROWS_PER_BLOCK][FBINS];  // async-staged imag row

    const int tid  = threadIdx.x;
    const int w    = tid >> 5;          // wave id == row-in-tile (0..15)
    const int lane = tid & 31;
    const int row  = blockIdx.x * ROWS_PER_BLOCK + w;
    const bool valid = row < nrows;
    const int bin0 = lane * 16;         // each lane owns 16 contiguous bins

    // ---------------- init LDS ----------------
    for (int i = tid; i < ROWS_PER_BLOCK * NBANDS; i += BLOCK_THREADS) {
        (&psumL[0][0])[i] = 0.f;
        (&lsumL[0][0])[i] = 0.f;
    }
    for (int i = tid; i < NBANDS * 32; i += BLOCK_THREADS) {
        int r = i >> 5, c = i & 31;
        sL[r][c] = (c < NBANDS) ? S[r * NBANDS + c] : 0.f;
    }
    {
        int bin = tid;                  // BLOCK_THREADS == FBINS
        int b = 0, hi = d_widths[0];
        while (bin >= hi) { ++b; hi += d_widths[b]; }
        segL[bin] = (unsigned char)b;
        float kb = (float)d_widths[b];
        float f = (bin + 1) * (32000.f / 1024.f / 1000.f);   // kHz
        float tq = 3.64f * __powf(f + 1e-6f, -0.8f)
                 - 6.5f * __expf(-0.6f * (f - 3.3f) * (f - 3.3f))
                 + 1e-3f * f * f * f * f;
        binTabL[bin] = make_float4(tq, 6.f / kb, 1.f / ((float)nrows * kb), (float)b);
    }
    if (tid < NBANDS) invkL[tid] = 1.f / (float)d_widths[tid];

    // ---------------- async-stage re/im rows into LDS (overlaps phases A/B) ----
    if (valid) {
        const uint32_t gbyte = (uint32_t)(((size_t)row * FBINS + bin0) * sizeof(float));
        const uint32_t ldsR  = (uint32_t)(uintptr_t)(&reL[w][bin0]);
        const uint32_t ldsI  = (uint32_t)(uintptr_t)(&imL[w][bin0]);
        asm volatile(
            "global_load_async_to_lds_b128 %0, %2, %3 offset:0\n\t"
            "global_load_async_to_lds_b128 %0, %2, %3 offset:16\n\t"
            "global_load_async_to_lds_b128 %0, %2, %3 offset:32\n\t"
            "global_load_async_to_lds_b128 %0, %2, %3 offset:48\n\t"
            "global_load_async_to_lds_b128 %1, %2, %4 offset:0\n\t"
            "global_load_async_to_lds_b128 %1, %2, %4 offset:16\n\t"
            "global_load_async_to_lds_b128 %1, %2, %4 offset:32\n\t"
            "global_load_async_to_lds_b128 %1, %2, %4 offset:48"
            :
            : "v"(ldsR), "v"(ldsI), "v"(gbyte),
              "s"((uint64_t)(uintptr_t)re), "s"((uint64_t)(uintptr_t)im)
            : "memory");
    }
    __syncthreads();

    // ---------------- phase A: spec + segmented band sums ----------------
    float spec[16];
    if (valid) {
        const float4* p = reinterpret_cast<const float4*>(lm + (size_t)row * FBINS + bin0);
        float v[16];
        #pragma unroll
        for (int q = 0; q < 4; ++q) {
            float4 x = p[q];
            v[4*q+0] = x.x; v[4*q+1] = x.y; v[4*q+2] = x.z; v[4*q+3] = x.w;
        }
        float runP = 0.f, runL = 0.f;
        int cur = segL[bin0];
        #pragma unroll
        for (int i = 0; i < 16; ++i) {
            float cl = fminf(fmaxf(v[i], -1000.f), 10.f);
            float s  = __expf(cl);
            spec[i] = s;
            int band = segL[bin0 + i];
            if (band != cur) {
                atomicAdd(&psumL[w][cur], runP);   // ds_add_f32, few per row
                atomicAdd(&lsumL[w][cur], runL);
                runP = 0.f; runL = 0.f; cur = band;
            }
            runP += s;
            runL += __log10f(s + 1e-30f);
        }
        atomicAdd(&psumL[w][cur], runP);
        atomicAdd(&lsumL[w][cur], runL);
    } else {
        #pragma unroll
        for (int i = 0; i < 16; ++i) spec[i] = 0.f;
    }
    __syncthreads();

    // ---------------- WMMA: spread = psum[16x24] @ S[24x24] (f32, K=4 steps) ----------------
    if (w < 2) {                                   // wave-uniform branch; EXEC all ones
        const int half = lane >> 4;                // 0: lanes 0-15, 1: lanes 16-31
        const int l16  = lane & 15;
        const int c0   = w * 16;                   // N-tile: bands 0-15 / 16-23(+pad)
        v8f acc = {};
        #pragma unroll
        for (int k0 = 0; k0 < NBANDS; k0 += 4) {
            v2f a, b;
            // A 16x4 layout: lanes16-31 hold K = k0+2/k0+3
            a.x = psumL[l16][k0 + 2 * half + 0];
            a.y = psumL[l16][k0 + 2 * half + 1];
            // B 4x16 layout: rows striped across lanes within each VGPR
            b.x = sL[k0 + 2 * half + 0][c0 + l16];
            b.y = sL[k0 + 2 * half + 1][c0 + l16];
            acc = __builtin_amdgcn_wmma_f32_16x16x4_f32(
                false, a, false, b, (short)0, acc, false, false);
        }
        // D 16x16: VGPR j -> rows j (lanes 0-15) / j+8 (lanes 16-31)
        #pragma unroll
        for (int j = 0; j < 8; ++j) {
            int m   = j + 8 * half;
            int col = c0 + l16;
            if (col < NBANDS) spreadL[m][col] = acc[j];
        }
    } else if (w == 2 && lane < NBANDS) {
        float s = 0.f;
        #pragma unroll
        for (int r = 0; r < NBANDS; ++r) s += sL[r][lane];
        ctestL[lane] = __log10f(s + 1e-8f);
    }
    __syncthreads();

    // ---------------- phase B: per-band masking threshold t ----------------
    if (lane < NBANDS) {
        float invk  = invkL[lane];
        float geo   = __expf((lsumL[w][lane] * invk) * 2.302585093f);  // 10^(logsum/k)
        float ari   = psumL[w][lane] * invk;
        float sfm   = 10.f * __log10f(geo * __builtin_amdgcn_rcpf(ari) + 1e-8f);
        float alpha = fminf(sfm * (-1.f / 60.f), 1.f);
        float off   = alpha * (14.5f + (float)(lane + 1)) + (1.f - alpha) * 5.5f;
        float toff  = -off * 0.1f;
        float c     = __log10f(spreadL[w][lane] + 1e-8f);
        float t     = __expf((c + toff) * 2.302585093f);
        float ren   = __expf((ctestL[lane] + toff) * 2.302585093f);
        tL[w][lane] = t * __builtin_amdgcn_rcpf(ren);
    }
    __syncthreads();

    // ---------------- phase C: perceptual entropy over bins ----------------
    asm volatile("s_wait_asynccnt 0x0" ::: "memory");   // re/im rows resident in LDS

    float acc = 0.f;
    if (valid) {
        const float4* pr = reinterpret_cast<const float4*>(&reL[w][bin0]);
        const float4* pi = reinterpret_cast<const float4*>(&imL[w][bin0]);
        float rv[16], iv[16];
        #pragma unroll
        for (int q = 0; q < 4; ++q) {
            float4 xr = pr[q], xi = pi[q];
            rv[4*q+0] = xr.x; rv[4*q+1] = xr.y; rv[4*q+2] = xr.z; rv[4*q+3] = xr.w;
            iv[4*q+0] = xi.x; iv[4*q+1] = xi.y; iv[4*q+2] = xi.z; iv[4*q+3] = xi.w;
        }
        #pragma unroll
        for (int i = 0; i < 16; ++i) {
            float4 tab = binTabL[bin0 + i];              // {tq, 6/k, wnorm/k, seg}
            int   band = (int)tab.w;
            float tb   = fmaxf(tL[w][band], tab.x);
            float invd = __builtin_amdgcn_rsqf(tb * tab.y);  // v_rsq_f32: 1/sqrt(6*tb/k)
            float sc   = 2.f * spec[i] * invd;
            float pe   = __log2f(fabsf(rv[i]) * sc + 1.f)
                       + __log2f(fabsf(iv[i]) * sc + 1.f);
            acc += pe * tab.z;
        }
    }

    // wave reduce, then block reduce -> per-block partial (deterministic)
    #pragma unroll
    for (int off = 16; off > 0; off >>= 1)
        acc += __shfl_down(acc, off, 32);
    if (lane == 0) waveAcc[w] = acc;
    __syncthreads();
    if (tid == 0) {
        float s = 0.f;
        #pragma unroll
        for (int i = 0; i < ROWS_PER_BLOCK; ++i) s += waveAcc[i];
        partials[blockIdx.x] = s;
    }
}

__global__ __launch_bounds__(256)
void pe_finalize(const float* __restrict__ partials, int n, float* __restrict__ out)
{
    __shared__ float buf[256];
    float s = 0.f;
    for (int i = threadIdx.x; i < n; i += 256) s += partials[i];
    buf[threadIdx.x] = s;
    __syncthreads();
    #pragma unroll
    for (int off = 128; off > 0; off >>= 1) {
        if (threadIdx.x < off) buf[threadIdx.x] += buf[threadIdx.x + off];
        __syncthreads();
    }
    if (threadIdx.x == 0) out[0] = 1.f / (buf[0] + 1.f);
}

extern "C" void kernel_launch(void* const* d_in, const int* in_sizes, int n_in,
                              void* d_out, int out_size, void* d_ws, size_t ws_size,
                              hipStream_t stream) {
    const float* lm = (const float*)d_in[0];   // log_magnitude [B,T,F]
    const float* re = (const float*)d_in[1];   // real          [B,T,F]
    const float* im = (const float*)d_in[2];   // imag          [B,T,F]
    const float* S  = (const float*)d_in[3];   // spread_function [NB,NB]

    int nrows   = in_sizes[0] / FBINS;                         // B*T = 38400
    int nblocks = (nrows + ROWS_PER_BLOCK - 1) / ROWS_PER_BLOCK;

    float* partials = (float*)d_ws;
    pe_main<<<nblocks, BLOCK_THREADS, 0, stream>>>(lm, re, im, S, partials, nrows);
    pe_finalize<<<1, 256, 0, stream>>>(partials, nblocks, (float*)d_out);
}